// NodeToEdge_82463372083407
// MI455X (gfx1250) — compile-verified
//
#include <hip/hip_runtime.h>
#include <hip/hip_bf16.h>

typedef __attribute__((ext_vector_type(2))) float v2f;
typedef __attribute__((ext_vector_type(8))) float v8f;

#define B_GRAPHS    48
#define N_NODES     128
#define F_FEAT      64
#define TWO_F       (2 * F_FEAT)                 // 128
#define P_PAIRS     (N_NODES * (N_NODES - 1))    // 16256
#define TOTAL_NODES (B_GRAPHS * N_NODES)         // 6144
#define BP          (B_GRAPHS * P_PAIRS)         // 780288

// ---------------------------------------------------------------------------
// Kernel 1: fold the two linear layers.
//   uv[f]      = sum_j W1[f, j]   * W2[j]   (f in 0..63  -> u,  applied to src)
//   uv[64 + f] = sum_j W1[64+f,j] * W2[j]   (              -> v,  applied to dst)
//   uv[128]    = b1 @ W2 + b2               (scalar bias k)
// 16K MACs total — one block, negligible.
// ---------------------------------------------------------------------------
__global__ void NodeToEdge_uv_kernel(const float* __restrict__ W1,
                                     const float* __restrict__ b1,
                                     const float* __restrict__ W2,
                                     const float* __restrict__ b2,
                                     float* __restrict__ uv) {
    const int f = threadIdx.x;  // 0..127
    float acc = 0.0f;
#pragma unroll 8
    for (int j = 0; j < TWO_F; ++j)
        acc += W1[f * TWO_F + j] * W2[j];
    uv[f] = acc;
    if (f == 0) {
        float k = b2[0];
        for (int j = 0; j < TWO_F; ++j)
            k += b1[j] * W2[j];
        uv[128] = k;
    }
}

// ---------------------------------------------------------------------------
// Kernel 2: per-node projection via V_WMMA_F32_16X16X4_F32.
// Computes S = X @ UV for X = node_feat [6144, 64] row-major and UV [64, 16]
// whose columns are {u, v, 0, ..., 0}. One wave owns a 16x16 output tile
// (16 node rows); K=64 is accumulated with 16 chained K=4 WMMAs.
//
// A layout (16x4 f32): lanes 0-15 hold M=lane, {v0,v1} = K0,K1;
//                      lanes 16-31 hold M=lane-16, {v0,v1} = K2,K3.
// B layout (4x16 f32): lanes 0-15 hold N=lane, {v0,v1} = K0,K1;
//                      lanes 16-31 hold N=lane-16, {v0,v1} = K2,K3.
// C/D layout:          VGPR r, lanes 0-15 -> S[r][lane];
//                      lanes 16-31 -> S[8+r][lane-16].
// Everything is branch-free up to the WMMAs so EXEC stays all-1s.
// ---------------------------------------------------------------------------
__global__ void NodeToEdge_project_kernel(const float* __restrict__ X,
                                          const float* __restrict__ uv,
                                          float* __restrict__ st) {
    const int lane = threadIdx.x & 31;
    const int wave = blockIdx.x * (blockDim.x >> 5) + (threadIdx.x >> 5);
    const int row0 = wave * 16;

    const int half = lane >> 4;   // 0: K pair {0,1}; 1: K pair {2,3}
    const int mn   = lane & 15;   // A-row / B-col within the tile
    const int col  = mn & 1;      // clamped UV column for safe loads

    const float* __restrict__ arow = X + (size_t)(row0 + mn) * F_FEAT;

    v8f c = {};
#pragma unroll
    for (int k0 = 0; k0 < F_FEAT; k0 += 4) {
        const int kk = k0 + 2 * half;

        v2f a;
        a.x = arow[kk];
        a.y = arow[kk + 1];

        // UV column mn: col0 = u = uv[0..63], col1 = v = uv[64..127], else 0.
        const float bx = uv[col * F_FEAT + kk];
        const float by = uv[col * F_FEAT + kk + 1];
        v2f b;
        b.x = (mn < 2) ? bx : 0.0f;   // v_cndmask, no EXEC change
        b.y = (mn < 2) ? by : 0.0f;

        c = __builtin_amdgcn_wmma_f32_16x16x4_f32(
                /*neg_a=*/false, a, /*neg_b=*/false, b,
                /*c_mod=*/(short)0, c, /*reuse_a=*/false, /*reuse_b=*/false);
    }

    // Lanes with N<2 hold the useful columns: N=0 -> s, N=1 -> t.
    if (mn < 2) {
#pragma unroll
        for (int r = 0; r < 8; ++r) {
            const int row = row0 + half * 8 + r;
            st[row * 2 + mn] = c[r];
        }
    }
}

// ---------------------------------------------------------------------------
// Kernel 3: emit edges. Bandwidth-bound: 3*B*P float stores (9.4 MB).
// Pair ordering matches itertools.permutations: p = i*(N-1) + q, j = q + (q>=i).
// Per-graph s/t vectors are staged in LDS (one 1 KB copy per block).
// Output: d_out[0..BP)=src ids, [BP..2BP)=dst ids, [2BP..3BP)=relu weights.
// ---------------------------------------------------------------------------
__global__ void NodeToEdge_edge_kernel(const float* __restrict__ st,
                                       const float* __restrict__ uv,
                                       float* __restrict__ out) {
    __shared__ float s_l[N_NODES];
    __shared__ float t_l[N_NODES];

    const int b   = blockIdx.y;
    const int tid = threadIdx.x;
    if (tid < N_NODES) {
        s_l[tid] = st[(b * N_NODES + tid) * 2 + 0];
        t_l[tid] = st[(b * N_NODES + tid) * 2 + 1];
    }
    __syncthreads();

    const float k = uv[128];
    const int p = blockIdx.x * blockDim.x + tid;
    if (p < P_PAIRS) {
        const int i = p / (N_NODES - 1);
        const int q = p - i * (N_NODES - 1);
        const int j = q + (q >= i ? 1 : 0);

        const float w = fmaxf(s_l[i] + t_l[j] + k, 0.0f);

        const int idx = b * P_PAIRS + p;
        out[idx]            = (float)(b * N_NODES + i);
        out[BP + idx]       = (float)(b * N_NODES + j);
        out[2 * BP + idx]   = w;
    }
}

// ---------------------------------------------------------------------------
// Inputs (setup_inputs order): 0=node_feat f32[6144*64], 1=batch_idx (unused),
// 2=n_graphs (unused, static 48), 3=W1 f32[128*128], 4=b1 f32[128],
// 5=W2 f32[128], 6=b2 f32[1].
// Workspace: uv (256-float slot incl. bias scalar) + st (2*6144 floats) ~50 KB.
// ---------------------------------------------------------------------------
extern "C" void kernel_launch(void* const* d_in, const int* in_sizes, int n_in,
                              void* d_out, int out_size, void* d_ws, size_t ws_size,
                              hipStream_t stream) {
    const float* X  = (const float*)d_in[0];
    const float* W1 = (const float*)d_in[3];
    const float* b1 = (const float*)d_in[4];
    const float* W2 = (const float*)d_in[5];
    const float* b2 = (const float*)d_in[6];

    float* uv = (float*)d_ws;      // [0..128] used, padded to 256
    float* st = uv + 256;          // [2 * TOTAL_NODES]

    NodeToEdge_uv_kernel<<<1, TWO_F, 0, stream>>>(W1, b1, W2, b2, uv);

    // 6144 rows / 16 rows per wave = 384 waves; 4 waves (128 thr) per block.
    NodeToEdge_project_kernel<<<TOTAL_NODES / 16 / 4, 128, 0, stream>>>(X, uv, st);

    dim3 egrid((P_PAIRS + 255) / 256, B_GRAPHS);
    NodeToEdge_edge_kernel<<<egrid, 256, 0, stream>>>(st, uv, (float*)d_out);
}